// MTSoftCrossEntropyLoss_53996328845913
// MI455X (gfx1250) — compile-verified
//
#include <hip/hip_runtime.h>
#include <stdint.h>

#define KDIM 32000
#define NROWS 4096
#define NTHREADS 320          // 10 wave32s
#define TILE 1280             // NTHREADS * 4 floats
#define NTILES 25             // 32000 / 1280 exactly
#define DEPTH 4               // async pipeline depth
#define NWAVES (NTHREADS / 32)

// Low 32 bits of a flat shared-aperture address == LDS byte offset
__device__ __forceinline__ unsigned lds_off_u32(const void* p) {
  return (unsigned)(uintptr_t)p;
}

// CDNA5 async DMA-style copy: 16 bytes per lane, global -> LDS, tracked by ASYNCcnt.
__device__ __forceinline__ void async_copy16(unsigned lds_addr, const float* base, unsigned byte_off) {
  asm volatile("global_load_async_to_lds_b128 %0, %1, %2"
               :: "v"(lds_addr), "v"(byte_off), "s"(base)
               : "memory");
}

__global__ __launch_bounds__(NTHREADS) void softce_rows(
    const float* __restrict__ xa, const float* __restrict__ ta,
    const float* __restrict__ xb, const float* __restrict__ tb,
    const float* __restrict__ w,  float* __restrict__ rowloss)
{
  __shared__ __align__(16) float xbuf[DEPTH][TILE];
  __shared__ __align__(16) float tbuf[DEPTH][TILE];
  __shared__ __align__(16) float wbuf[DEPTH][TILE];
  __shared__ float redm[NWAVES], redz[NWAVES], reds1[NWAVES], reds2[NWAVES];

  const int tid  = threadIdx.x;
  const int blk  = blockIdx.x;
  const int task = blk >> 12;            // 0: task a, 1: task b
  const int row  = blk & (NROWS - 1);

  const float* xrow = (task ? xb : xa) + (size_t)row * KDIM;  // uniform -> SGPR base
  const float* trow = (task ? tb : ta) + (size_t)row * KDIM;

  const unsigned lane16 = (unsigned)tid * 16u;  // per-thread byte slot inside a tile

  auto issue = [&](int tile) {
    const int buf = tile % DEPTH;
    const unsigned goff = (unsigned)(tile * TILE * 4) + lane16;   // byte column offset
    async_copy16(lds_off_u32(&xbuf[buf][0]) + lane16, xrow, goff);
    async_copy16(lds_off_u32(&tbuf[buf][0]) + lane16, trow, goff);
    async_copy16(lds_off_u32(&wbuf[buf][0]) + lane16, w,    goff);
  };

  // Prologue: fill DEPTH-1 tiles of the pipeline.
  issue(0); issue(1); issue(2);

  float m  = -__builtin_inff();
  float z  = 0.f;
  float s1 = 0.f;   // sum t*w*x
  float s2 = 0.f;   // sum t*w

  for (int i = 0; i < NTILES; ++i) {
    const int nb = i + DEPTH - 1;
    // Uniform (SALU) branch: keep exactly the right number of async ops in flight.
    if (nb < NTILES) {
      issue(nb);
      asm volatile("s_wait_asynccnt 9" ::: "memory");   // tiles i+1..i+3 outstanding
    } else if (i == NTILES - 3) {
      asm volatile("s_wait_asynccnt 6" ::: "memory");   // tiles i+1, i+2 outstanding
    } else if (i == NTILES - 2) {
      asm volatile("s_wait_asynccnt 3" ::: "memory");   // tile i+1 outstanding
    } else {
      asm volatile("s_wait_asynccnt 0" ::: "memory");   // drain
    }

    const int buf = i % DEPTH;
    const float4 xv = *(const float4*)&xbuf[buf][tid * 4];
    const float4 tv = *(const float4*)&tbuf[buf][tid * 4];
    const float4 wv = *(const float4*)&wbuf[buf][tid * 4];

    // Chunk-wise online softmax: one rescale exp per 4 elements (1.25 exp/elem).
    const float tm = fmaxf(fmaxf(xv.x, xv.y), fmaxf(xv.z, xv.w));
    const float nm = fmaxf(m, tm);
    const float e0 = __expf(xv.x - nm);
    const float e1 = __expf(xv.y - nm);
    const float e2 = __expf(xv.z - nm);
    const float e3 = __expf(xv.w - nm);
    z = fmaf(z, __expf(m - nm), (e0 + e1) + (e2 + e3));
    m = nm;

    const float tw0 = tv.x * wv.x;
    const float tw1 = tv.y * wv.y;
    const float tw2 = tv.z * wv.z;
    const float tw3 = tv.w * wv.w;
    s1 = fmaf(tw0, xv.x, s1);
    s1 = fmaf(tw1, xv.y, s1);
    s1 = fmaf(tw2, xv.z, s1);
    s1 = fmaf(tw3, xv.w, s1);
    s2 += (tw0 + tw1) + (tw2 + tw3);
  }

  // Wave32 tree reduction with (m, z) rescale-merge.
  #pragma unroll
  for (int off = 16; off > 0; off >>= 1) {
    const float mo  = __shfl_xor(m,  off, 32);
    const float zo  = __shfl_xor(z,  off, 32);
    const float s1o = __shfl_xor(s1, off, 32);
    const float s2o = __shfl_xor(s2, off, 32);
    const float M = fmaxf(m, mo);
    z = z * __expf(m - M) + zo * __expf(mo - M);
    m = M; s1 += s1o; s2 += s2o;
  }

  const int wave = tid >> 5;
  if ((tid & 31) == 0) {
    redm[wave] = m; redz[wave] = z; reds1[wave] = s1; reds2[wave] = s2;
  }
  __syncthreads();

  if (tid == 0) {
    float M = redm[0], Z = redz[0], S1 = reds1[0], S2 = reds2[0];
    #pragma unroll
    for (int i2 = 1; i2 < NWAVES; ++i2) {
      const float mo = redm[i2];
      const float Mx = fmaxf(M, mo);
      Z = Z * __expf(M - Mx) + redz[i2] * __expf(mo - Mx);
      M = Mx; S1 += reds1[i2]; S2 += reds2[i2];
    }
    const float lse = M + __logf(Z);
    rowloss[blk] = lse * S2 - S1;     // -sum t*w*logp for this row
  }
}

// Deterministic fixed-order final reduction of 8192 row losses (no fp atomics).
__global__ __launch_bounds__(256) void reduce_rows(const float* __restrict__ rowloss,
                                                   float* __restrict__ out)
{
  __shared__ float s[256];
  float acc = 0.f;
  #pragma unroll
  for (int j = 0; j < (2 * NROWS) / 256; ++j)
    acc += rowloss[threadIdx.x + j * 256];
  s[threadIdx.x] = acc;
  __syncthreads();
  for (int off = 128; off > 0; off >>= 1) {
    if (threadIdx.x < off) s[threadIdx.x] += s[threadIdx.x + off];
    __syncthreads();
  }
  if (threadIdx.x == 0) out[0] = s[0] * (1.0f / (float)NROWS);  // mean_a + mean_b = grand_sum / N
}

extern "C" void kernel_launch(void* const* d_in, const int* in_sizes, int n_in,
                              void* d_out, int out_size, void* d_ws, size_t ws_size,
                              hipStream_t stream) {
  const float* xa = (const float*)d_in[0];
  const float* ta = (const float*)d_in[1];
  const float* xb = (const float*)d_in[2];
  const float* tb = (const float*)d_in[3];
  const float* w  = (const float*)d_in[4];
  float* rowloss  = (float*)d_ws;          // 2*NROWS floats = 32 KB scratch

  softce_rows<<<2 * NROWS, NTHREADS, 0, stream>>>(xa, ta, xb, tb, w, rowloss);
  reduce_rows<<<1, 256, 0, stream>>>(rowloss, (float*)d_out);
}